// MultKAN_20590073217462
// MI455X (gfx1250) — compile-verified
//
#include <hip/hip_runtime.h>
#include <math.h>

typedef __attribute__((ext_vector_type(2))) float v2f;
typedef __attribute__((ext_vector_type(8))) float v8f;

// ---------------------------------------------------------------------------
// Problem constants (from reference)
// ---------------------------------------------------------------------------
#define BATCH     2048
#define IN0       128      // W_IN[0]
#define OUT0      96       // W_OUT[1]
#define IN1       80       // W_IN[1]
#define OUT1      32       // W_OUT[2]
#define NB        9        // 8 spline basis funcs + 1 silu channel
#define K0        (IN0 * NB)   // 1152
#define K1        (IN1 * NB)   // 720
#define DIMSUM0   64       // WIDTH[1][0]

// ---------------------------------------------------------------------------
// Effective-weight prep: W[(i*9+g), j] from coef/scales/mask, s = j*in+i
// ---------------------------------------------------------------------------
__global__ void kan_prep_w(const float* __restrict__ coef,
                           const float* __restrict__ sb,
                           const float* __restrict__ sp,
                           const float* __restrict__ mask,
                           float* __restrict__ Wt,
                           int in_dim, int out_dim, int size) {
    int s = blockIdx.x * blockDim.x + threadIdx.x;
    if (s >= size) return;
    int i = s % in_dim;
    int j = s / in_dim;
    float mk  = mask[s];
    float msp = mk * sp[s];
    size_t kbase = (size_t)i * NB;
#pragma unroll
    for (int g = 0; g < 8; ++g)
        Wt[(kbase + g) * out_dim + j] = msp * coef[(size_t)s * 8 + g];
    Wt[(kbase + 8) * out_dim + j] = mk * sb[s];
}

// symbolic layer constant per output column: Σ_i mask_s[j,i]*affine[j,i,3]
__global__ void kan_prep_symb(const float* __restrict__ sm,
                              const float* __restrict__ sa,
                              float* __restrict__ symb,
                              int in_dim, int out_dim) {
    int j = blockIdx.x * blockDim.x + threadIdx.x;
    if (j >= out_dim) return;
    float acc = 0.0f;
    for (int i = 0; i < in_dim; ++i) {
        size_t e = (size_t)j * in_dim + i;
        acc += sm[e] * sa[e * 4 + 3];
    }
    symb[j] = acc;
}

// ---------------------------------------------------------------------------
// Feature kernel: F[b, i*9+g] = B_g(x[b,i]) for g<8, silu(x[b,i]) for g==8.
// Cubic Cox–de Boor on 12 extended knots, matching the reference recursion
// (half-open interval indicator at order 0).
// ---------------------------------------------------------------------------
__global__ void kan_features(const float* __restrict__ X,
                             const float* __restrict__ grid, // row 0: 6 knots
                             float* __restrict__ F,
                             int in_dim, int total) {
    int t = blockIdx.x * blockDim.x + threadIdx.x;
    if (t >= total) return;
    int b = t / in_dim;
    int i = t - b * in_dim;

    float g0 = grid[0], g5 = grid[5];
    float h = (g5 - g0) * 0.2f;       // (grid[-1]-grid[0]) / (npts-1)
    float gk[12];
#pragma unroll
    for (int q = 0; q < 6; ++q) gk[3 + q] = grid[q];
    gk[2] = g0 - h;  gk[1] = g0 - 2.0f * h;  gk[0] = g0 - 3.0f * h;
    gk[9] = g5 + h;  gk[10] = g5 + 2.0f * h; gk[11] = g5 + 3.0f * h;

    float x = X[t];

    float v[11];
#pragma unroll
    for (int q = 0; q < 11; ++q)
        v[q] = (x >= gk[q] && x < gk[q + 1]) ? 1.0f : 0.0f;
#pragma unroll
    for (int p = 1; p <= 3; ++p) {
#pragma unroll
        for (int q = 0; q <= 10 - p; ++q) {
            float num0 = (x - gk[q]) / (gk[q + p] - gk[q]);
            float num1 = (gk[q + p + 1] - x) / (gk[q + p + 1] - gk[q + 1]);
            v[q] = num0 * v[q] + num1 * v[q + 1];
        }
    }

    size_t K = (size_t)in_dim * NB;
    float* o = F + (size_t)b * K + (size_t)i * NB;
#pragma unroll
    for (int g = 0; g < 8; ++g) o[g] = v[g];
    o[8] = x / (1.0f + expf(-x));     // silu
}

// ---------------------------------------------------------------------------
// WMMA f32 GEMM: Y[m0:16, n0:16] = A[M,K] x Bw[K,N] (+ symb[n]).
// One wave (32 lanes) per 16x16 tile; V_WMMA_F32_16X16X4_F32, K step 4.
// A frag layout (ISA 7.12.2): lane=(h,m): a0=A[m,k+2h], a1=A[m,k+2h+1].
// C/D: VGPR r -> row r+8h, col m.
// ---------------------------------------------------------------------------
__global__ void __launch_bounds__(32)
kan_gemm_wmma(const float* __restrict__ A,
              const float* __restrict__ Bw,
              const float* __restrict__ symb,
              float* __restrict__ Y,
              int K, int N) {
    const int m0   = blockIdx.x << 4;
    const int n0   = blockIdx.y << 4;
    const int lane = threadIdx.x;
    const int hh   = lane >> 4;       // 0 / 1 half
    const int m    = lane & 15;

    const float* arow = A + (size_t)(m0 + m) * K + 2 * hh;
    const float* bcol = Bw + (size_t)(2 * hh) * N + (n0 + m);
    const size_t bstep = (size_t)N;

    v8f c = {};
    for (int k = 0; k < K; k += 4) {
        v2f a, b;
        a.x = arow[k];
        a.y = arow[k + 1];
        b.x = bcol[(size_t)k * bstep];
        b.y = bcol[(size_t)(k + 1) * bstep];
        c = __builtin_amdgcn_wmma_f32_16x16x4_f32(
                /*neg_a=*/false, a, /*neg_b=*/false, b,
                /*c_mod=*/(short)0, c, /*reuse_a=*/false, /*reuse_b=*/false);
    }

    float sj = symb[n0 + m];
#pragma unroll
    for (int r = 0; r < 8; ++r) {
        int row = m0 + r + 8 * hh;
        Y[(size_t)row * N + n0 + m] = c[r] + sj;
    }
}

// ---------------------------------------------------------------------------
// Inter-layer fusion: x1[:, :64]=y[:, :64]; x1[:, 64+p]=y[:,64+2p]*y[:,65+2p];
// then + bias0. (dim_sum=64, 16 product pairs -> width 80)
// ---------------------------------------------------------------------------
__global__ void kan_fuse0(const float* __restrict__ Y,
                          const float* __restrict__ bias,
                          float* __restrict__ X1) {
    int t = blockIdx.x * blockDim.x + threadIdx.x;
    if (t >= BATCH * IN1) return;
    int b = t / IN1;
    int c = t - b * IN1;
    const float* yr = Y + (size_t)b * OUT0;
    float v;
    if (c < DIMSUM0) {
        v = yr[c];
    } else {
        int p = c - DIMSUM0;
        v = yr[DIMSUM0 + 2 * p] * yr[DIMSUM0 + 2 * p + 1];
    }
    X1[t] = v + bias[c];
}

// Final: out = y1 (already includes symb1) + bias1   (dim_sum=32 -> no pairs)
__global__ void kan_fuse1(const float* __restrict__ Y,
                          const float* __restrict__ bias,
                          float* __restrict__ Out) {
    int t = blockIdx.x * blockDim.x + threadIdx.x;
    if (t >= BATCH * OUT1) return;
    int j = t & (OUT1 - 1);
    Out[t] = Y[t] + bias[j];
}

// ---------------------------------------------------------------------------
// Launch
// ---------------------------------------------------------------------------
extern "C" void kernel_launch(void* const* d_in, const int* in_sizes, int n_in,
                              void* d_out, int out_size, void* d_ws, size_t ws_size,
                              hipStream_t stream) {
    (void)in_sizes; (void)n_in; (void)out_size; (void)ws_size;

    const float* x        = (const float*)d_in[0];
    const float* grid0    = (const float*)d_in[1];
    const float* coef0    = (const float*)d_in[2];
    const float* sbase0   = (const float*)d_in[3];
    const float* ssp0     = (const float*)d_in[4];
    const float* mask0    = (const float*)d_in[5];
    const float* smask0   = (const float*)d_in[6];
    const float* saff0    = (const float*)d_in[7];
    const float* bias0    = (const float*)d_in[8];
    const float* grid1    = (const float*)d_in[9];
    const float* coef1    = (const float*)d_in[10];
    const float* sbase1   = (const float*)d_in[11];
    const float* ssp1     = (const float*)d_in[12];
    const float* mask1    = (const float*)d_in[13];
    const float* smask1   = (const float*)d_in[14];
    const float* saff1    = (const float*)d_in[15];
    const float* bias1    = (const float*)d_in[16];
    float* out = (float*)d_out;

    // workspace layout (floats), F/Y regions shared across layers
    float* ws = (float*)d_ws;
    float* W0 = ws;                       // K0*OUT0 = 110592
    float* W1 = W0 + (size_t)K0 * OUT0;   // K1*OUT1 = 23040
    float* S0 = W1 + (size_t)K1 * OUT1;   // 96
    float* S1 = S0 + OUT0;                // 32
    float* Fb = S1 + OUT1;                // BATCH*K0 = 2359296 (max of layers)
    float* Yb = Fb + (size_t)BATCH * K0;  // BATCH*OUT0 = 196608 (max of layers)
    float* X1 = Yb + (size_t)BATCH * OUT0;// BATCH*IN1 = 163840

    // ---- layer 0 ----
    kan_prep_w   <<<(IN0 * OUT0 + 255) / 256, 256, 0, stream>>>(coef0, sbase0, ssp0, mask0, W0, IN0, OUT0, IN0 * OUT0);
    kan_prep_symb<<<1, OUT0, 0, stream>>>(smask0, saff0, S0, IN0, OUT0);
    kan_features <<<(BATCH * IN0 + 255) / 256, 256, 0, stream>>>(x, grid0, Fb, IN0, BATCH * IN0);
    kan_gemm_wmma<<<dim3(BATCH / 16, OUT0 / 16), 32, 0, stream>>>(Fb, W0, S0, Yb, K0, OUT0);
    kan_fuse0    <<<(BATCH * IN1 + 255) / 256, 256, 0, stream>>>(Yb, bias0, X1);

    // ---- layer 1 ----
    kan_prep_w   <<<(IN1 * OUT1 + 255) / 256, 256, 0, stream>>>(coef1, sbase1, ssp1, mask1, W1, IN1, OUT1, IN1 * OUT1);
    kan_prep_symb<<<1, OUT1, 0, stream>>>(smask1, saff1, S1, IN1, OUT1);
    kan_features <<<(BATCH * IN1 + 255) / 256, 256, 0, stream>>>(X1, grid1, Fb, IN1, BATCH * IN1);
    kan_gemm_wmma<<<dim3(BATCH / 16, OUT1 / 16), 32, 0, stream>>>(Fb, W1, S1, Yb, K1, OUT1);
    kan_fuse1    <<<(BATCH * OUT1 + 255) / 256, 256, 0, stream>>>(Yb, bias1, out);
}